// WaveletDecompReconstruct_DWT_20143396618891
// MI455X (gfx1250) — compile-verified
//
#include <hip/hip_runtime.h>
#include <math.h>

// CDNA5 / gfx1250 fused 3-level DWT analysis+synthesis.
//
// Structure: kernel-2 stride-2 filters over 3 levels => transform is block
// diagonal over groups of 8 samples: rec8 = M(8x8) @ x8.  We form
// Mp = diag(M, M) (rows 0-7 real, rows 8-15 imag of one block) and evaluate
// Y = Mp @ X with 4 chained V_WMMA_F32_16X16X4_F32 (K = 16 in chunks of 4),
// 16 blocks (columns) per wave-iteration.
//
// Bandwidth-bound: 256 MB min traffic -> ~11 us @ 23.3 TB/s; the 8 MAC/elem
// transform would need ~47 TFLOP/s f32 at full rate, so it is routed through
// the f32 WMMA pipe instead of scalar VALU FMAs.

typedef __attribute__((ext_vector_type(2))) float v2f;
typedef __attribute__((ext_vector_type(4))) float v4f;
typedef __attribute__((ext_vector_type(8))) float v8f;

#define SEQ   65536
#define BATCH 256
#define NTHR  512            // 16 waves of 32
#define NWAVE (NTHR / 32)
#define NTILE (SEQ / 128)    // 16 blocks of 8 per tile -> 512 tiles per row

__global__ __launch_bounds__(NTHR)
void dwt_wmma_fused(const float* __restrict__ xr, const float* __restrict__ xi,
                    const float* __restrict__ wlo, const float* __restrict__ whi,
                    float* __restrict__ outr, float* __restrict__ outi)
{
    __shared__ float sred[3 * NTHR];
    __shared__ float lM[64];        // composite 8x8 matrix, row-major
    __shared__ float stats[4];      // mean_r, mean_i, std, inv_std

    const int tid = threadIdx.x;
    const size_t rowoff = (size_t)blockIdx.x * SEQ;
    const float* pr = xr + rowoff;
    const float* pi = xi + rowoff;

    // ---- Build M by pushing basis vectors through analysis+synthesis ----
    if (tid == 0) {
        float wl[3][2], wh[3][2];
        #pragma unroll
        for (int j = 0; j < 3; ++j) {
            wl[j][0] = wlo[2 * j]; wl[j][1] = wlo[2 * j + 1];
            wh[j][0] = whi[2 * j]; wh[j][1] = whi[2 * j + 1];
        }
        #pragma unroll
        for (int col = 0; col < 8; ++col) {
            float v[8];
            #pragma unroll
            for (int i = 0; i < 8; ++i) v[i] = (i == col) ? 1.0f : 0.0f;
            float l1[4], H1[4];
            #pragma unroll
            for (int n = 0; n < 4; ++n) {
                l1[n] = wl[0][0] * v[2 * n] + wl[0][1] * v[2 * n + 1];
                H1[n] = wh[0][0] * v[2 * n] + wh[0][1] * v[2 * n + 1];
            }
            float l2[2], H2[2];
            #pragma unroll
            for (int n = 0; n < 2; ++n) {
                l2[n] = wl[1][0] * l1[2 * n] + wl[1][1] * l1[2 * n + 1];
                H2[n] = wh[1][0] * l1[2 * n] + wh[1][1] * l1[2 * n + 1];
            }
            float l3 = wl[2][0] * l2[0] + wl[2][1] * l2[1];
            float H3 = wh[2][0] * l2[0] + wh[2][1] * l2[1];
            float r2[2];
            #pragma unroll
            for (int k = 0; k < 2; ++k) r2[k] = wl[2][k] * l3 + wh[2][k] * H3;
            float r1[4];
            #pragma unroll
            for (int n = 0; n < 2; ++n)
                #pragma unroll
                for (int k = 0; k < 2; ++k)
                    r1[2 * n + k] = wl[1][k] * r2[n] + wh[1][k] * H2[n];
            #pragma unroll
            for (int n = 0; n < 4; ++n)
                #pragma unroll
                for (int k = 0; k < 2; ++k)
                    lM[(2 * n + k) * 8 + col] = wl[0][k] * r1[n] + wh[0][k] * H1[n];
        }
    }

    // ---- Pass 1: mean / variance (single streaming pass, f32 accum) ----
    float sr = 0.0f, si = 0.0f, sq = 0.0f;
    for (int it = tid * 4; it < SEQ; it += NTHR * 4) {
        v4f r = *(const v4f*)(pr + it);
        v4f m = *(const v4f*)(pi + it);
        sr += r.x + r.y + r.z + r.w;
        si += m.x + m.y + m.z + m.w;
        sq += r.x * r.x + r.y * r.y + r.z * r.z + r.w * r.w
            + m.x * m.x + m.y * m.y + m.z * m.z + m.w * m.w;
    }
    sred[tid] = sr; sred[NTHR + tid] = si; sred[2 * NTHR + tid] = sq;
    __syncthreads();
    for (int s = NTHR / 2; s > 0; s >>= 1) {
        if (tid < s) {
            sred[tid]            += sred[tid + s];
            sred[NTHR + tid]     += sred[NTHR + tid + s];
            sred[2 * NTHR + tid] += sred[2 * NTHR + tid + s];
        }
        __syncthreads();
    }
    if (tid == 0) {
        const float inv = 1.0f / (float)SEQ;
        float mr = sred[0] * inv;
        float mi = sred[NTHR] * inv;
        float var = 0.5f * (sred[2 * NTHR] * inv - mr * mr - mi * mi) + 1e-5f;
        float sd = sqrtf(var);
        stats[0] = mr; stats[1] = mi; stats[2] = sd; stats[3] = 1.0f / sd;
    }
    __syncthreads();

    const float mean_r = stats[0], mean_i = stats[1];
    const float sd = stats[2], invsd = stats[3];

    const int lane = tid & 31;
    const int wave = tid >> 5;
    const int col  = lane & 15;      // block instance within tile / A row
    const bool hi  = lane >= 16;

    // ---- A fragments: Mp = diag(M, M), 16x4 layout per ISA 7.12.2 ----
    // lane L holds row (L%16); VGPR0 = K c, VGPR1 = K c+1, c = 4k + (hi?2:0)
    v2f A[4];
    #pragma unroll
    for (int k = 0; k < 4; ++k) {
        int c = 4 * k + (hi ? 2 : 0);
        float a0, a1;
        if (c < 8) {            // columns 0-7: only real rows (i<8) nonzero
            a0 = (col < 8) ? lM[col * 8 + c]     : 0.0f;
            a1 = (col < 8) ? lM[col * 8 + c + 1] : 0.0f;
        } else {                // columns 8-15: only imag rows (i>=8) nonzero
            a0 = (col >= 8) ? lM[(col - 8) * 8 + (c - 8)] : 0.0f;
            a1 = (col >= 8) ? lM[(col - 8) * 8 + (c - 7)] : 0.0f;
        }
        A[k] = (v2f){a0, a1};
    }

    const float mean_sel = hi ? mean_i : mean_r;
    float* po = (hi ? outi : outr) + rowoff;

    // ---- Pass 2: streamed WMMA transform, 128 samples/component per tile ----
    for (int t = wave; t < NTILE; t += NWAVE) {   // uniform trip count per wave
        const size_t eb  = (size_t)128 * t + 8 * (size_t)col;   // block base
        const size_t ib  = eb + (hi ? 2 : 0);
        // B fragments (4x16 per chunk): chunks 0/1 from real, 2/3 from imag
        v2f r01 = *(const v2f*)(pr + ib);
        v2f r45 = *(const v2f*)(pr + ib + 4);
        v2f i01 = *(const v2f*)(pi + ib);
        v2f i45 = *(const v2f*)(pi + ib + 4);
        v2f B0 = (r01 - mean_r) * invsd;
        v2f B1 = (r45 - mean_r) * invsd;
        v2f B2 = (i01 - mean_i) * invsd;
        v2f B3 = (i45 - mean_i) * invsd;

        v8f acc = {0.f, 0.f, 0.f, 0.f, 0.f, 0.f, 0.f, 0.f};
        acc = __builtin_amdgcn_wmma_f32_16x16x4_f32(false, A[0], false, B0,
                                                    (short)0, acc, false, false);
        acc = __builtin_amdgcn_wmma_f32_16x16x4_f32(false, A[1], false, B1,
                                                    (short)0, acc, false, false);
        acc = __builtin_amdgcn_wmma_f32_16x16x4_f32(false, A[2], false, B2,
                                                    (short)0, acc, false, false);
        acc = __builtin_amdgcn_wmma_f32_16x16x4_f32(false, A[3], false, B3,
                                                    (short)0, acc, false, false);

        // D: lane L, vgpr v = feature v (+8 if hi) of block col -> 8
        // contiguous floats per lane; denormalize and stream out (NT stores).
        v4f o0 = { fmaf(acc[0], sd, mean_sel), fmaf(acc[1], sd, mean_sel),
                   fmaf(acc[2], sd, mean_sel), fmaf(acc[3], sd, mean_sel) };
        v4f o1 = { fmaf(acc[4], sd, mean_sel), fmaf(acc[5], sd, mean_sel),
                   fmaf(acc[6], sd, mean_sel), fmaf(acc[7], sd, mean_sel) };
        __builtin_nontemporal_store(o0, (v4f*)(po + eb));
        __builtin_nontemporal_store(o1, (v4f*)(po + eb + 4));
    }
}

extern "C" void kernel_launch(void* const* d_in, const int* in_sizes, int n_in,
                              void* d_out, int out_size, void* d_ws, size_t ws_size,
                              hipStream_t stream)
{
    (void)in_sizes; (void)n_in; (void)out_size; (void)d_ws; (void)ws_size;
    const float* xr  = (const float*)d_in[0];
    const float* xi  = (const float*)d_in[1];
    const float* wlo = (const float*)d_in[2];
    const float* whi = (const float*)d_in[3];
    float* outr = (float*)d_out;
    float* outi = outr + (size_t)BATCH * SEQ;
    dwt_wmma_fused<<<BATCH, NTHR, 0, stream>>>(xr, xi, wlo, whi, outr, outi);
}